// SSIMLoss_19172734009817
// MI455X (gfx1250) — compile-verified
//
#include <hip/hip_runtime.h>

typedef float v2f __attribute__((ext_vector_type(2)));
typedef float v8f __attribute__((ext_vector_type(8)));

#define H_IMG   512
#define W_IMG   512
#define TILE    32
#define HALO    5
#define INP     48            // padded halo buffer: 48x48 (valid 42x42, rest zero)
#define HROWS   48            // horizontal-conv result rows (3 chunks of 16)
#define HP      36            // H buffer pitch (32 cols + pad)
#define HFSZ    (HROWS * HP)  // one H field

// D = A(16x4 f32) * B(4x16 f32) + C(16x16 f32), wave32, full-precision WMMA
__device__ __forceinline__ v8f wmma4(v2f a, v2f b, v8f c) {
  return __builtin_amdgcn_wmma_f32_16x16x4_f32(false, a, false, b, (short)0, c,
                                               false, false);
}

#if defined(__gfx1250__) && __has_builtin(__builtin_amdgcn_global_load_async_to_lds_b32)
#define USE_ASYNC_LDS 1
typedef __attribute__((address_space(1))) int* g32_ptr;   // global (AS1) src
typedef __attribute__((address_space(3))) int* l32_ptr;   // LDS (AS3) dst
#else
#define USE_ASYNC_LDS 0
#endif

__global__ __launch_bounds__(128) void ssim_tile_kernel(
    const float* __restrict__ x, const float* __restrict__ y,
    float* __restrict__ gsum) {
  __shared__ float sX[INP * INP];
  __shared__ float sY[INP * INP];
  __shared__ float sH[5 * HFSZ];
  __shared__ float wgsum;

  const int tid  = threadIdx.x;
  const int lane = tid & 31;
  const int wave = tid >> 5;
  const int hi   = lane >> 4;   // lane half selects K pair
  const int nm   = lane & 15;   // M (A rows) / N (B,C cols)

  const int plane = blockIdx.y;
  const int tx    = blockIdx.x & 15;
  const int ty    = blockIdx.x >> 4;
  const int row0  = ty * TILE;
  const int col0  = tx * TILE;
  const float* px = x + (size_t)plane * (H_IMG * W_IMG);
  const float* py = y + (size_t)plane * (H_IMG * W_IMG);

  // Per-lane banded Gaussian fragments: wfrag[c] = W[4c+2*hi+{0,1}][nm],
  // W[p][n] = g[p-n] (11-tap, sigma=1.5). Same layout serves the H-pass B
  // operand and the V-pass A operand.
  v2f wfrag[7];
#pragma unroll
  for (int c2 = 0; c2 < 7; ++c2) {
#pragma unroll
    for (int e = 0; e < 2; ++e) {
      const int d   = 4 * c2 + 2 * hi + e - nm;
      const float t = (float)(d - 5);
      const float v = __expf(t * t * (-1.0f / 4.5f)) * (1.0f / 3.75923274f);
      wfrag[c2][e] = ((unsigned)d < 11u) ? v : 0.0f;
    }
  }
  if (tid == 0) wgsum = 0.0f;

  // ---- Stage 1: zero-padded halo -> LDS (42x42 valid inside 48x48) ----
  for (int idx = tid; idx < INP * INP; idx += 128) {
    const int r  = idx / INP;
    const int c  = idx - r * INP;
    const int gr = row0 - HALO + r;
    const int gc = col0 - HALO + c;
    const bool ok = (r < TILE + 2 * HALO) && (c < TILE + 2 * HALO) &&
                    (gr >= 0) && (gr < H_IMG) && (gc >= 0) && (gc < W_IMG);
    const size_t go = (size_t)gr * W_IMG + gc;
#if USE_ASYNC_LDS
    if (ok) {
      __builtin_amdgcn_global_load_async_to_lds_b32((g32_ptr)(px + go),
                                                    (l32_ptr)(&sX[idx]), 0, 0);
      __builtin_amdgcn_global_load_async_to_lds_b32((g32_ptr)(py + go),
                                                    (l32_ptr)(&sY[idx]), 0, 0);
    } else {
      sX[idx] = 0.0f;
      sY[idx] = 0.0f;
    }
#else
    if (ok) __builtin_prefetch(&px[go], 0, 0);   // global_prefetch_b8
    sX[idx] = ok ? px[go] : 0.0f;
    sY[idx] = ok ? py[go] : 0.0f;
#endif
  }
#if USE_ASYNC_LDS
#if __has_builtin(__builtin_amdgcn_s_wait_asynccnt)
  __builtin_amdgcn_s_wait_asynccnt(0);
#else
  asm volatile("s_wait_asynccnt 0x0" ::: "memory");
#endif
#endif
  __syncthreads();

  // ---- Stage 2: horizontal 11-tap conv via WMMA, all 5 fields per subtile ----
  // subtile st = rowchunk(3) x coltile(2); H[r][j] = sum_k g[k]*in[r][j+k]
  for (int st = wave; st < 6; st += 4) {
    const int rbase = (st >> 1) * 16;
    const int j0    = (st & 1) * 16;
    v8f aX = {}, aY = {}, aXX = {}, aYY = {}, aXY = {};
#pragma unroll
    for (int c2 = 0; c2 < 7; ++c2) {
      const int col = j0 + 4 * c2 + 2 * hi;              // even -> 8B aligned
      const int off = (rbase + nm) * INP + col;
      const v2f xv = *reinterpret_cast<const v2f*>(&sX[off]);
      const v2f yv = *reinterpret_cast<const v2f*>(&sY[off]);
      const v2f w  = wfrag[c2];
      aX  = wmma4(xv, w, aX);          // 5 independent accumulation chains:
      aY  = wmma4(yv, w, aY);          // no WMMA->WMMA RAW inside one chunk
      aXX = wmma4(xv * xv, w, aXX);
      aYY = wmma4(yv * yv, w, aYY);
      aXY = wmma4(xv * yv, w, aXY);
    }
    const int so = (rbase + 8 * hi) * HP + j0 + nm;
#pragma unroll
    for (int v = 0; v < 8; ++v) {
      sH[0 * HFSZ + so + v * HP] = aX[v];
      sH[1 * HFSZ + so + v * HP] = aY[v];
      sH[2 * HFSZ + so + v * HP] = aXX[v];
      sH[3 * HFSZ + so + v * HP] = aYY[v];
      sH[4 * HFSZ + so + v * HP] = aXY[v];
    }
  }
  __syncthreads();

  // ---- Stage 3: vertical 11-tap conv via WMMA; one 16x16 subtile per wave ----
  const int m0 = (wave >> 1) * 16;   // output row base within tile
  const int n0 = (wave & 1) * 16;    // output col base within tile
  v8f accf[5];
#pragma unroll
  for (int f = 0; f < 5; ++f) {
    const float* Hf = &sH[f * HFSZ];
    v8f acc = {};
#pragma unroll
    for (int c2 = 0; c2 < 7; ++c2) {
      const int pr = m0 + 4 * c2 + 2 * hi;
      v2f b;
      b.x = Hf[pr * HP + n0 + nm];
      b.y = Hf[(pr + 1) * HP + n0 + nm];
      acc = wmma4(wfrag[c2], b, acc);
    }
    accf[f] = acc;
  }

  // ---- Pointwise SSIM + reduction ----
  const float C1v = 1e-4f, C2v = 9e-4f;
  float s = 0.0f;
#pragma unroll
  for (int v = 0; v < 8; ++v) {
    const float mu1 = accf[0][v], mu2 = accf[1][v];
    const float ex2 = accf[2][v], ey2 = accf[3][v], exy = accf[4][v];
    const float m11 = mu1 * mu1, m22 = mu2 * mu2, m12 = mu1 * mu2;
    const float s1 = ex2 - m11, s2 = ey2 - m22, s12 = exy - m12;
    const float num = (2.0f * m12 + C1v) * (2.0f * s12 + C2v);
    const float den = (m11 + m22 + C1v) * (s1 + s2 + C2v);
    s += num / den;
  }
#pragma unroll
  for (int off = 16; off > 0; off >>= 1) s += __shfl_xor(s, off, 32);
  if (lane == 0) atomicAdd(&wgsum, s);
  __syncthreads();
  if (tid == 0) atomicAdd(gsum, wgsum);
}

__global__ void ssim_finalize_kernel(const float* __restrict__ gsum,
                                     float* __restrict__ out) {
  out[0] = 1.0f - gsum[0] * (1.0f / 12582912.0f);  // N = 16*3*512*512
}

extern "C" void kernel_launch(void* const* d_in, const int* in_sizes, int n_in,
                              void* d_out, int out_size, void* d_ws,
                              size_t ws_size, hipStream_t stream) {
  const float* img1 = (const float*)d_in[0];
  const float* img2 = (const float*)d_in[1];
  float* gsum = (float*)d_ws;
  (void)hipMemsetAsync(gsum, 0, sizeof(float), stream);
  dim3 grid((W_IMG / TILE) * (H_IMG / TILE), 48);  // 256 tiles x 48 planes
  ssim_tile_kernel<<<grid, 128, 0, stream>>>(img1, img2, gsum);
  ssim_finalize_kernel<<<1, 1, 0, stream>>>(gsum, (float*)d_out);
}